// VAE_Encoder_58291296141649
// MI455X (gfx1250) — compile-verified
//
#include <hip/hip_runtime.h>

typedef float v2f __attribute__((ext_vector_type(2)));
typedef float v8f __attribute__((ext_vector_type(8)));

#define B_SZ 512
#define T_SZ 2048
#define D_SZ 64
#define H1_SZ 80
#define U_SZ 65
#define NU4 260      // 4*U
#define D2_SZ 13
#define L_SZ 8
#define GP 80        // padded per-gate width
#define NP 320       // padded 4U
#define KXP 64       // K of input GEMM
#define KHP 80       // padded K of recurrence GEMM
#define XSTRIDE 68   // LDS row stride for x tile (16B aligned, conflict-free)
#define HSTRIDE 82   // LDS row stride for h tile (8B aligned, conflict-free)
#define MT 16        // batch tile (WMMA M)

// Fast gate nonlinearities: single-instruction TRANS ops (v_exp_f32 / v_rcp_f32)
// instead of IEEE-divide sequences and ocml tanh polynomials. These sit on the
// sequential recurrence critical path, so latency here matters more than ulps.
__device__ __forceinline__ float fast_sigmoid(float x) {
  return __builtin_amdgcn_rcpf(1.f + __expf(-x));
}
__device__ __forceinline__ float fast_tanh(float x) {
  // tanh(x) = 1 - 2/(1 + exp(2x))
  return fmaf(-2.f, __builtin_amdgcn_rcpf(1.f + __expf(2.f * x)), 1.f);
}

// ---- Prep: fold Dense(80) into LSTM input projection; pad weights ----
__global__ void prep_wc_kernel(const float* __restrict__ W1, const float* __restrict__ b1,
                               const float* __restrict__ Wk, const float* __restrict__ bl,
                               float* __restrict__ Wc, float* __restrict__ bc) {
  int idx = blockIdx.x * blockDim.x + threadIdx.x;
  int stride = gridDim.x * blockDim.x;
  for (int e = idx; e < D_SZ * NP; e += stride) {
    int d = e / NP, c = e % NP;
    int q = c / GP, u = c % GP;
    float s = 0.f;
    if (u < U_SZ) {
      int src = q * U_SZ + u;
      for (int j = 0; j < H1_SZ; ++j) s += W1[d * H1_SZ + j] * Wk[j * NU4 + src];
    }
    Wc[e] = s;
  }
  for (int c = idx; c < NP; c += stride) {
    int q = c / GP, u = c % GP;
    float s = 0.f;
    if (u < U_SZ) {
      int src = q * U_SZ + u;
      s = bl[src];
      for (int j = 0; j < H1_SZ; ++j) s += b1[j] * Wk[j * NU4 + src];
    }
    bc[c] = s;
  }
}

__global__ void prep_wr_kernel(const float* __restrict__ Wr, float* __restrict__ Wrp) {
  int idx = blockIdx.x * blockDim.x + threadIdx.x;
  int stride = gridDim.x * blockDim.x;
  for (int e = idx; e < KHP * NP; e += stride) {
    int k = e / NP, c = e % NP;
    int q = c / GP, u = c % GP;
    Wrp[e] = (k < U_SZ && u < U_SZ) ? Wr[k * NU4 + q * U_SZ + u] : 0.f;
  }
}

// ---- Fused: input projection + 2048-step LSTM + dense tails ----
// Block = 16 batch rows, 5 waves; wave w owns u in [16w,16w+16) for all 4 gates.
__global__ __launch_bounds__(160, 1)
void vae_lstm_fused(const float* __restrict__ x, const float* __restrict__ eps,
                    const float* __restrict__ Wc, const float* __restrict__ bc,
                    const float* __restrict__ Wrp,
                    const float* __restrict__ W2, const float* __restrict__ b2,
                    const float* __restrict__ Wm, const float* __restrict__ bm,
                    const float* __restrict__ Wv, const float* __restrict__ bvv,
                    float* __restrict__ out) {
  extern __shared__ float smem[];
  float* wc_s = smem;                      // [KXP/2][NP] float2-swizzled (80 KB)
  float* wr_s = wc_s + KXP * NP;           // [KHP/2][NP] float2-swizzled (100 KB)
  float* bc_s = wr_s + KHP * NP;           // [NP]
  float* xb   = bc_s + NP;                 // 2 x [MT][XSTRIDE] double buffer
  float* hb   = xb + 2 * MT * XSTRIDE;     // [MT][HSTRIDE]

  const int tid  = threadIdx.x;
  const int lane = tid & 31;
  const int l16  = lane & 15;              // M-row for A frag, N-col for B/C/D frags
  const int hi   = lane >> 4;              // lane half: K-pair select / M+8 select
  const int wave = tid >> 5;
  const int u0   = wave * 16;
  const int b0   = blockIdx.x * MT;

  // Stage weights into LDS, swizzled as (k/2, n) -> {W[k][n], W[k+1][n]} so each
  // B-fragment is one conflict-free ds_load_b64.
  for (int e = tid; e < (KXP / 2) * NP; e += 160) {
    int kp = e / NP, n = e - kp * NP;
    wc_s[2 * e]     = Wc[(2 * kp) * NP + n];
    wc_s[2 * e + 1] = Wc[(2 * kp + 1) * NP + n];
  }
  for (int e = tid; e < (KHP / 2) * NP; e += 160) {
    int kp = e / NP, n = e - kp * NP;
    wr_s[2 * e]     = Wrp[(2 * kp) * NP + n];
    wr_s[2 * e + 1] = Wrp[(2 * kp + 1) * NP + n];
  }
  for (int e = tid; e < NP; e += 160) bc_s[e] = bc[e];
  for (int e = tid; e < MT * HSTRIDE; e += 160) hb[e] = 0.f;

  // Stage x_0
  const int xr = tid >> 3, xs = (tid & 7) * 8;
  if (tid < 128) {
    const float* gp = x + ((size_t)(b0 + xr) * T_SZ) * D_SZ + xs;
    float4 q0 = *(const float4*)gp;
    float4 q1 = *(const float4*)(gp + 4);
    float* xd = xb + xr * XSTRIDE + xs;
    *(float4*)(xd) = q0;
    *(float4*)(xd + 4) = q1;
  }
  __syncthreads();

  // Loop-invariant gate biases (one per gate, per lane's output column).
  float bias[4];
#pragma unroll
  for (int q = 0; q < 4; ++q) bias[q] = bc_s[q * GP + u0 + l16];

  v8f c_state;
#pragma unroll
  for (int r = 0; r < 8; ++r) c_state[r] = 0.f;

  for (int t = 0; t < T_SZ; ++t) {
    // Prefetch x_{t+1} into registers (overlaps with this step's WMMA work).
    float4 p0, p1;
    const bool pf = (t + 1 < T_SZ) && (tid < 128);
    if (pf) {
      const float* gp = x + ((size_t)(b0 + xr) * T_SZ + (t + 1)) * D_SZ + xs;
      p0 = *(const float4*)gp;
      p1 = *(const float4*)(gp + 4);
    }

    // Init gate accumulators with folded bias (value depends only on column).
    v8f acc[4];
#pragma unroll
    for (int q = 0; q < 4; ++q) {
      v8f a;
#pragma unroll
      for (int r = 0; r < 8; ++r) a[r] = bias[q];
      acc[q] = a;
    }

    // GEMM1: z += x_t @ Wc   (K=64, fp32 WMMA, 4 independent gate chains)
    const float* xcur = xb + (t & 1) * MT * XSTRIDE;
#pragma unroll
    for (int k = 0; k < KXP; k += 4) {
      v2f av = *(const v2f*)(xcur + l16 * XSTRIDE + k + 2 * hi);
#pragma unroll
      for (int q = 0; q < 4; ++q) {
        v2f bfr = *(const v2f*)(wc_s + (((k >> 1) + hi) * NP + (q * GP + u0 + l16)) * 2);
        acc[q] = __builtin_amdgcn_wmma_f32_16x16x4_f32(false, av, false, bfr,
                                                       (short)0, acc[q], false, false);
      }
    }
    // GEMM2: z += h_{t-1} @ Wr   (K=80 padded)
#pragma unroll
    for (int k = 0; k < KHP; k += 4) {
      v2f av = *(const v2f*)(hb + l16 * HSTRIDE + k + 2 * hi);
#pragma unroll
      for (int q = 0; q < 4; ++q) {
        v2f bfr = *(const v2f*)(wr_s + (((k >> 1) + hi) * NP + (q * GP + u0 + l16)) * 2);
        acc[q] = __builtin_amdgcn_wmma_f32_16x16x4_f32(false, av, false, bfr,
                                                       (short)0, acc[q], false, false);
      }
    }

    // Gates (c lives in registers in exact C/D fragment layout).
    v8f hnew;
#pragma unroll
    for (int r = 0; r < 8; ++r) {
      float ig = fast_sigmoid(acc[0][r]);
      float fg = fast_sigmoid(acc[1][r]);
      float gg = fast_tanh(acc[2][r]);
      float og = fast_sigmoid(acc[3][r]);
      float cn = fmaf(fg, c_state[r], ig * gg);
      c_state[r] = cn;
      hnew[r] = og * fast_tanh(cn);
    }

    __syncthreads();  // all waves done reading hb / xb for step t
#pragma unroll
    for (int r = 0; r < 8; ++r)
      hb[(r + 8 * hi) * HSTRIDE + u0 + l16] = hnew[r];
    if (pf) {
      float* xd = xb + ((t + 1) & 1) * MT * XSTRIDE + xr * XSTRIDE + xs;
      *(float4*)(xd) = p0;
      *(float4*)(xd + 4) = p1;
    }
    __syncthreads();  // hb / next x buffer visible
  }

  // Tails: fc2 = h@W2+b2 ; heads ; reparameterize. Tiny — scalar per (row, l).
  if (tid < 128) {
    const int r = tid >> 3, l = tid & 7;
    float fc2[D2_SZ];
#pragma unroll
    for (int k = 0; k < D2_SZ; ++k) {
      float s = b2[k];
      for (int u = 0; u < U_SZ; ++u) s += hb[r * HSTRIDE + u] * W2[u * D2_SZ + k];
      fc2[k] = s;
    }
    float zm = bm[l], zv = bvv[l];
#pragma unroll
    for (int k = 0; k < D2_SZ; ++k) {
      zm += fc2[k] * Wm[k * L_SZ + l];
      zv += fc2[k] * Wv[k * L_SZ + l];
    }
    const int b = b0 + r;
    float e = eps[b * L_SZ + l];
    float z = zm + sqrtf(__expf(zv)) * e;
    out[b * L_SZ + l] = zm;
    out[B_SZ * L_SZ + b * L_SZ + l] = zv;
    out[2 * B_SZ * L_SZ + b * L_SZ + l] = z;
  }
}

extern "C" void kernel_launch(void* const* d_in, const int* in_sizes, int n_in,
                              void* d_out, int out_size, void* d_ws, size_t ws_size,
                              hipStream_t stream) {
  const float* x   = (const float*)d_in[0];
  const float* eps = (const float*)d_in[1];
  const float* W1  = (const float*)d_in[2];
  const float* b1  = (const float*)d_in[3];
  const float* Wk  = (const float*)d_in[4];
  const float* Wr  = (const float*)d_in[5];
  const float* bl  = (const float*)d_in[6];
  const float* W2  = (const float*)d_in[7];
  const float* b2  = (const float*)d_in[8];
  const float* Wm  = (const float*)d_in[9];
  const float* bm  = (const float*)d_in[10];
  const float* Wv  = (const float*)d_in[11];
  const float* bv  = (const float*)d_in[12];
  float* out = (float*)d_out;
  float* ws  = (float*)d_ws;

  float* Wc  = ws;                 // [64][320]
  float* bc  = Wc + D_SZ * NP;     // [320]
  float* Wrp = bc + NP;            // [80][320]

  prep_wc_kernel<<<80, 256, 0, stream>>>(W1, b1, Wk, bl, Wc, bc);
  prep_wr_kernel<<<100, 256, 0, stream>>>(Wr, Wrp);

  const size_t smem =
      (size_t)(KXP * NP + KHP * NP + NP + 2 * MT * XSTRIDE + MT * HSTRIDE) * sizeof(float);
  hipFuncSetAttribute(reinterpret_cast<const void*>(vae_lstm_fused),
                      hipFuncAttributeMaxDynamicSharedMemorySize, (int)smem);
  vae_lstm_fused<<<dim3(B_SZ / MT), dim3(160), smem, stream>>>(
      x, eps, Wc, bc, Wrp, W2, b2, Wm, bm, Wv, bv, out);

  (void)in_sizes; (void)n_in; (void)out_size; (void)ws_size;
}